// Pinn_54838142435792
// MI455X (gfx1250) — compile-verified
//
#include <hip/hip_runtime.h>
#include <hip/hip_bf16.h>

typedef __attribute__((ext_vector_type(2))) float v2f;
typedef __attribute__((ext_vector_type(8))) float v8f;
typedef __attribute__((ext_vector_type(4))) unsigned int u32x4;
typedef __attribute__((ext_vector_type(4))) int i32x4;
typedef __attribute__((ext_vector_type(8))) int i32x8;

#define HID   20          // hidden width
#define NJET  20          // 3rd-order jet components in 3 vars: 1+3+6+10
#define PTSW  4           // points per wave
#define ROWS  (PTSW*NJET) // 80 rows = 5 exact 16-row M-tiles
#define ASTR  20          // LDS row stride (floats)
#define WAVES 4
#define BLOCK (WAVES*32)
#define NHID  7           // number of 20x20 hidden matmuls (W_h)

// LDS layout (floats)
#define WI_OFF 0                  // W_in [3][20]
#define BI_OFF 60                 // b_in [20]
#define WH_OFF 80                 // W_h  [7][20][20]  (TDM-staged)
#define BH_OFF 2880               // b_h  [7][20]
#define WO_OFF 3020               // W_out[20][2]
#define BO_OFF 3060               // b_out[2]
#define WAVE_OFF 3062             // per-wave act/pre regions
#define WAVE_FLOATS (2*ROWS*ASTR) // 3200
#define SMEM_FLOATS (WAVE_OFF + WAVES*WAVE_FLOATS) // 15862 -> 63448 B

#define WH_FLOATS 2800

__device__ __forceinline__ int idx2(int a, int b) { // a<=b, pairs over 3 vars
    return (a == 0) ? b : ((a == 1) ? (2 + b) : 5);
}

// Propagate a 3rd-order (3-variable) jet through tanh.
__device__ __forceinline__ void tanh_jet(const float* s, float* o) {
    float t  = tanhf(s[0]);
    float d1 = 1.f - t * t;                 // f'
    float d2 = -2.f * t * d1;               // f''
    float d3 = -2.f * d1 * (1.f - 3.f*t*t); // f'''
    o[0] = t;
#pragma unroll
    for (int a = 0; a < 3; ++a) o[1 + a] = d1 * s[1 + a];
    const int A2[6] = {0,0,0,1,1,2}, B2[6] = {0,1,2,1,2,2};
#pragma unroll
    for (int i = 0; i < 6; ++i) {
        float sa = s[1 + A2[i]], sb = s[1 + B2[i]];
        o[4 + i] = d1 * s[4 + i] + d2 * sa * sb;
    }
    const int A3[10] = {0,0,0,0,0,0,1,1,1,2};
    const int B3[10] = {0,0,0,1,1,2,1,1,2,2};
    const int C3[10] = {0,1,2,1,2,2,1,2,2,2};
#pragma unroll
    for (int i = 0; i < 10; ++i) {
        int a = A3[i], b = B3[i], c = C3[i];
        float sa = s[1+a], sb = s[1+b], sc = s[1+c];
        float sab = s[4+idx2(a,b)], sac = s[4+idx2(a,c)], sbc = s[4+idx2(b,c)];
        o[10 + i] = d1*s[10+i] + d2*(sab*sc + sac*sb + sbc*sa) + d3*sa*sb*sc;
    }
}

__global__ void pinn_zero_loss(float* out, int Btot) {
    if (threadIdx.x == 0 && blockIdx.x == 0) out[(size_t)3 * Btot] = 0.f;
}

__global__ __launch_bounds__(BLOCK)
void pinn_jet_wmma(const float* __restrict__ x,  const float* __restrict__ y,
                   const float* __restrict__ tt, const float* __restrict__ u,
                   const float* __restrict__ v,
                   const float* __restrict__ Wi, const float* __restrict__ bi,
                   const float* __restrict__ Wh, const float* __restrict__ bh,
                   const float* __restrict__ Wo, const float* __restrict__ bo,
                   const float* __restrict__ l1p, const float* __restrict__ l2p,
                   float* __restrict__ out, int Btot) {
    extern __shared__ float smem[];
    const int tid = threadIdx.x;

    // ---- stage weights into LDS once per block ----
    // W_h (11.2 KB) goes through the Tensor Data Mover as a 1-D descriptor
    // copy issued by wave 0 (tracked by TENSORcnt); the small arrays go
    // through normal loads on all threads in parallel.
#if defined(__HIP_DEVICE_COMPILE__) && __has_builtin(__builtin_amdgcn_tensor_load_to_lds)
    if (tid < 32) {
        unsigned long long ga = (unsigned long long)(const void*)Wh;
        u32x4 g0;
        g0.x = 1u;                                   // count=1, user descriptor
        g0.y = (unsigned)(WH_OFF * 4);               // lds_addr (bytes)
        g0.z = (unsigned)(ga & 0xffffffffu);         // global_addr[31:0]
        g0.w = (unsigned)((ga >> 32) & 0x1ffffffu)   // global_addr[56:32]
             | (2u << 30);                           // type = 2 ("image")
        i32x8 g1;
        g1[0] = (int)(2u << 16);                     // data_size = 4 bytes
        g1[1] = (int)((WH_FLOATS & 0xffff) << 16);   // tensor_dim0[15:0]
        g1[2] = (int)(1u << 16);                     // tensor_dim1 = 1
        g1[3] = (int)((WH_FLOATS & 0xffff) << 16);   // tile_dim0 = 2800
        g1[4] = 1;                                   // tile_dim1 = 1
        g1[5] = WH_FLOATS;                           // tensor_dim0_stride
        g1[6] = 0; g1[7] = 0;
        i32x4 z4 = {0, 0, 0, 0};
#if __clang_major__ >= 23
        i32x8 z8 = {0, 0, 0, 0, 0, 0, 0, 0};
        __builtin_amdgcn_tensor_load_to_lds(g0, g1, z4, z4, z8, 0);
#else
        __builtin_amdgcn_tensor_load_to_lds(g0, g1, z4, z4, 0);
#endif
        __builtin_amdgcn_s_wait_tensorcnt(0);
    }
#else
    for (int i = tid; i < WH_FLOATS; i += BLOCK) smem[WH_OFF + i] = Wh[i];
#endif
    for (int i = tid; i < 60;   i += BLOCK) smem[WI_OFF + i] = Wi[i];
    for (int i = tid; i < 20;   i += BLOCK) smem[BI_OFF + i] = bi[i];
    for (int i = tid; i < 140;  i += BLOCK) smem[BH_OFF + i] = bh[i];
    for (int i = tid; i < 40;   i += BLOCK) smem[WO_OFF + i] = Wo[i];
    for (int i = tid; i < 2;    i += BLOCK) smem[BO_OFF + i] = bo[i];
    __syncthreads();

    const int lane = tid & 31, wave = tid >> 5;
    const int hi = lane >> 4, lcol = lane & 15;
    float* wAct = smem + WAVE_OFF + wave * WAVE_FLOATS; // [80][ASTR] jets of activations
    float* wPre = wAct + ROWS * ASTR;                   // [80][ASTR] pre-activation jets
    const int base = blockIdx.x * (WAVES * PTSW) + wave * PTSW;

    // ---- input layer (K=3, done in VALU): pre-activation jets, then tanh jet ----
    for (int it = lane; it < PTSW * HID; it += 32) {
        int p = it / HID, n = it % HID;
        int gi = min(base + p, Btot - 1);
        float s[NJET], o[NJET];
        float xx = x[gi], yy = y[gi], tz = tt[gi];
        float w0 = smem[WI_OFF + 0*HID + n];
        float w1 = smem[WI_OFF + 1*HID + n];
        float w2 = smem[WI_OFF + 2*HID + n];
        s[0] = xx*w0 + yy*w1 + tz*w2 + smem[BI_OFF + n];
        s[1] = w0; s[2] = w1; s[3] = w2;
#pragma unroll
        for (int j = 4; j < NJET; ++j) s[j] = 0.f;
        tanh_jet(s, o);
#pragma unroll
        for (int j = 0; j < NJET; ++j) wAct[(p*NJET + j)*ASTR + n] = o[j];
    }
    __syncthreads();

    // ---- 7 hidden layers: 80x20x20 jet GEMM via V_WMMA_F32_16X16X4_F32 ----
    for (int l = 0; l < NHID; ++l) {
        // B operand (W_h[l], K x N): VGPR0 = K {0,2} by lane half, VGPR1 = K {1,3}
        v2f breg[2][5];
#pragma unroll
        for (int nt = 0; nt < 2; ++nt)
#pragma unroll
            for (int ks = 0; ks < 5; ++ks) {
                int col = nt*16 + lcol;
                int kb  = ks*4 + hi*2;
                float b0 = 0.f, b1 = 0.f;
                if (col < HID) {
                    b0 = smem[WH_OFF + (l*HID + kb    )*HID + col];
                    b1 = smem[WH_OFF + (l*HID + kb + 1)*HID + col];
                }
                v2f bb = {b0, b1};
                breg[nt][ks] = bb;
            }
        for (int mt = 0; mt < 5; ++mt) {
            // A operand (jets x k): lane = row, VGPR0 = K {0,2}, VGPR1 = K {1,3}
            v2f areg[5];
#pragma unroll
            for (int ks = 0; ks < 5; ++ks) {
                int row = mt*16 + lcol;
                int kb  = ks*4 + hi*2;
                v2f aa = { wAct[row*ASTR + kb], wAct[row*ASTR + kb + 1] };
                areg[ks] = aa;
            }
#pragma unroll
            for (int nt = 0; nt < 2; ++nt) {
                v8f acc = {};
#pragma unroll
                for (int ks = 0; ks < 5; ++ks)
                    acc = __builtin_amdgcn_wmma_f32_16x16x4_f32(
                        false, areg[ks], false, breg[nt][ks],
                        (short)0, acc, false, false);
                int col = nt*16 + lcol;
                if (col < HID) {
#pragma unroll
                    for (int r = 0; r < 8; ++r) {
                        int grow = mt*16 + hi*8 + r;          // C/D: M = r + 8*laneHalf
                        float val = acc[r];
                        if ((grow % NJET) == 0)               // bias only on value row
                            val += smem[BH_OFF + l*HID + col];
                        wPre[grow*ASTR + col] = val;
                    }
                }
            }
        }
        __syncthreads();
        // tanh jet nonlinearity per (point, neuron)
        for (int it = lane; it < PTSW * HID; it += 32) {
            int p = it / HID, n = it % HID;
            float s[NJET], o[NJET];
#pragma unroll
            for (int j = 0; j < NJET; ++j) s[j] = wPre[(p*NJET + j)*ASTR + n];
            tanh_jet(s, o);
#pragma unroll
            for (int j = 0; j < NJET; ++j) wAct[(p*NJET + j)*ASTR + n] = o[j];
        }
        __syncthreads();
    }

    // ---- output layer (N=2): psi / p jets per row, VALU ----
    for (int row = lane; row < ROWS; row += 32) {
        int j = row % NJET;
        float ps = 0.f, pr = 0.f;
#pragma unroll
        for (int k = 0; k < HID; ++k) {
            float a = wAct[row*ASTR + k];
            ps += a * smem[WO_OFF + k*2 + 0];
            pr += a * smem[WO_OFF + k*2 + 1];
        }
        if (j == 0) { ps += smem[BO_OFF + 0]; pr += smem[BO_OFF + 1]; }
        wPre[row*ASTR + 0] = ps;
        wPre[row*ASTR + 1] = pr;
    }
    __syncthreads();

    // ---- extract derivatives, residuals, preds, loss ----
    float lossLocal = 0.f;
    if (lane < PTSW) {
        int p = lane, gi = base + p;
        if (gi < Btot) {
            float l1 = l1p[0], l2 = l2p[0];
            #define PJ(j) (wPre[(p*NJET + (j))*ASTR + 0])
            #define QJ(j) (wPre[(p*NJET + (j))*ASTR + 1])
            float u_pred =  PJ(2);          // psi_y
            float v_pred = -PJ(1);          // -psi_x
            float u_x =  PJ(5),  u_y =  PJ(7),  u_t =  PJ(8);
            float v_x = -PJ(4),  v_y = -PJ(5),  v_t = -PJ(6);
            float u_xx =  PJ(11), u_yy =  PJ(16);
            float v_xx = -PJ(10), v_yy = -PJ(13);
            float p_pred = QJ(0), p_x = QJ(1), p_y = QJ(2);
            #undef PJ
            #undef QJ
            float f_u = l1*(u_t + u_pred*u_x + v_pred*u_y) + p_x - l2*(u_xx + u_yy);
            float f_v = l1*(v_t + u_pred*v_x + v_pred*v_y) - l1*9.81f
                        + p_y - l2*(v_xx + v_yy);
            out[gi*3 + 0] = p_pred;
            out[gi*3 + 1] = u_pred;
            out[gi*3 + 2] = v_pred;
            float du = u[gi] - u_pred, dv = v[gi] - v_pred;
            lossLocal = du*du + dv*dv + f_u*f_u + f_v*f_v;
        }
    }
#pragma unroll
    for (int off = 16; off > 0; off >>= 1)
        lossLocal += __shfl_down(lossLocal, off, 32);
    if (lane == 0) atomicAdd(&out[(size_t)3 * Btot], lossLocal);
}

extern "C" void kernel_launch(void* const* d_in, const int* in_sizes, int n_in,
                              void* d_out, int out_size, void* d_ws, size_t ws_size,
                              hipStream_t stream) {
    const float* x  = (const float*)d_in[0];
    const float* y  = (const float*)d_in[1];
    const float* t  = (const float*)d_in[2];
    const float* u  = (const float*)d_in[3];
    const float* v  = (const float*)d_in[4];
    const float* Wi = (const float*)d_in[5];
    const float* bi = (const float*)d_in[6];
    const float* Wh = (const float*)d_in[7];
    const float* bh = (const float*)d_in[8];
    const float* Wo = (const float*)d_in[9];
    const float* bo = (const float*)d_in[10];
    const float* l1 = (const float*)d_in[11];
    const float* l2 = (const float*)d_in[12];
    float* out = (float*)d_out;
    int Btot = in_sizes[0];

    pinn_zero_loss<<<1, 32, 0, stream>>>(out, Btot);

    int ptsPerBlock = WAVES * PTSW; // 16
    int grid = (Btot + ptsPerBlock - 1) / ptsPerBlock;
    size_t shmem = (size_t)SMEM_FLOATS * sizeof(float); // 63448 B < 64 KB
    pinn_jet_wmma<<<grid, BLOCK, shmem, stream>>>(
        x, y, t, u, v, Wi, bi, Wh, bh, Wo, bo, l1, l2, out, Btot);
}